// LocalAttention_54039278518803
// MI455X (gfx1250) — compile-verified
//
#include <hip/hip_runtime.h>
#include <hip/hip_bf16.h>

// ---------------------------------------------------------------------------
// Local windowed attention, MI455X (gfx1250).
//   B=8, S=4096, D=512, H=8, hd=64, WINDOW=5, STRIDE=4 -> P=4100, nw=1024
// Pipeline:
//   0) init: zero accA/attn_weights, bias-fill padded row p=4096 of Q/K/V,
//      repack weights into K-paired float2 layout
//   1) fused QKV projection GEMM over B*S rows, fp32 WMMA 16x16x4,
//      uniform K-loop (no divergence), immediate-offset epilogue
//   2) per-window 5x5 attention (LDS); atomics only on window-boundary rows
//   3) output GEMM accA @ Wo with counts/bias epilogue, fp32 WMMA 16x16x4
// ---------------------------------------------------------------------------

#define BATCH 8
#define SEQ   4096
#define DIM   512
#define PADP  4100                    // S + pad
#define NW    1024                    // windows per batch
#define NH    8
#define HD    64
#define MROWS (BATCH * SEQ)           // 32768 GEMM rows = 1024 * 32
#define PROWS (BATCH * PADP)          // 32800 padded rows (Q/K/V/accA)
#define NE    ((size_t)PROWS * DIM)   // elements per padded activation buffer
#define KPAIR (DIM / 2)               // 256 K-pairs
#define WPACK (KPAIR * DIM)           // 131072 float2 per packed weight

typedef float v2f __attribute__((ext_vector_type(2)));
typedef float v8f __attribute__((ext_vector_type(8)));

__device__ __forceinline__ v8f wmma_f32(v2f a, v2f b, v8f c) {
  // D = A(16x4) * B(4x16) + C(16x16), all fp32  ->  v_wmma_f32_16x16x4_f32
  return __builtin_amdgcn_wmma_f32_16x16x4_f32(
      /*neg_a=*/false, a, /*neg_b=*/false, b,
      /*c_mod=*/(short)0, c, /*reuse_a=*/false, /*reuse_b=*/false);
}

// ---------------------------------------------------------------------------
// 0) init: zero accA + attnW, bias-fill padded Q/K/V row, pack weights.
//    Packed layout: Wp[kp * DIM + n] = { W[2kp][n], W[2kp+1][n] }.
// ---------------------------------------------------------------------------
__global__ __launch_bounds__(256) void init_kernel(
    float* __restrict__ accA, float* __restrict__ attnW,
    float* __restrict__ Q, float* __restrict__ K, float* __restrict__ V,
    const float* __restrict__ bq, const float* __restrict__ bk,
    const float* __restrict__ bv,
    const float* __restrict__ Wq, const float* __restrict__ Wk,
    const float* __restrict__ Wv, const float* __restrict__ Wo,
    v2f* __restrict__ Wqp, v2f* __restrict__ Wkp,
    v2f* __restrict__ Wvp, v2f* __restrict__ Wop) {
  for (size_t i = (size_t)blockIdx.x * blockDim.x + threadIdx.x; i < NE;
       i += (size_t)gridDim.x * blockDim.x) {
    accA[i] = 0.0f;
    if (i < (size_t)(BATCH * NH * 5 * 5)) attnW[i] = 0.0f;
    if (i < (size_t)(BATCH * DIM)) {  // padded row p=4096 of each batch
      const int b = (int)(i >> 9);
      const int c = (int)(i & (DIM - 1));
      const size_t o = ((size_t)b * PADP + SEQ) * DIM + c;
      Q[o] = bq[c];
      K[o] = bk[c];
      V[o] = bv[c];
    }
    if (i < (size_t)(4 * WPACK)) {
      const int m = (int)(i / WPACK);
      const int j = (int)(i - (size_t)m * WPACK);
      const int kp = j >> 9;           // K pair
      const int n  = j & (DIM - 1);    // column
      const float* W = (m == 0) ? Wq : (m == 1) ? Wk : (m == 2) ? Wv : Wo;
      v2f* Wp        = (m == 0) ? Wqp : (m == 1) ? Wkp : (m == 2) ? Wvp : Wop;
      v2f pr = { W[(size_t)(2 * kp) * DIM + n],
                 W[(size_t)(2 * kp + 1) * DIM + n] };
      Wp[j] = pr;
    }
  }
}

// ---------------------------------------------------------------------------
// 1) fused QKV projection over the real B*S rows.
//    One wave owns a 32x32 C tile for each of Q,K,V (12 accumulators).
//    grid = (MROWS/32 = 1024, DIM/128 = 4), block = 128 (4 waves).
// ---------------------------------------------------------------------------
__global__ __launch_bounds__(128) void gemm_qkv_kernel(
    const float* __restrict__ x,
    const v2f* __restrict__ Wqp, const float* __restrict__ bq,
    const v2f* __restrict__ Wkp, const float* __restrict__ bk,
    const v2f* __restrict__ Wvp, const float* __restrict__ bv,
    float* __restrict__ Q, float* __restrict__ K, float* __restrict__ V) {
  const int lane = threadIdx.x & 31;
  const int wave = threadIdx.x >> 5;
  const int half = lane >> 4;     // selects K pair (0,1) vs (2,3) in the frag
  const int l    = lane & 15;     // M row (A) / N col (B,C,D) within tile

  const int m0 = blockIdx.x * 32;
  const int n0 = blockIdx.y * 128 + wave * 32;

  // per-lane bases; loop index below is UNIFORM so the loop unrolls and the
  // addresses fold into base + immediate offsets
  const v2f* A0 = (const v2f*)(x + (size_t)(m0 + l) * DIM) + half;
  const v2f* A1 = (const v2f*)(x + (size_t)(m0 + 16 + l) * DIM) + half;
  const int c0 = n0 + l;
  const int c1 = n0 + 16 + l;
  const int w0 = (half << 9) + c0;  // per-lane weight base (pair half + col)
  const int w1 = (half << 9) + c1;

  v8f q00 = {}, q01 = {}, q10 = {}, q11 = {};
  v8f k00 = {}, k01 = {}, k10 = {}, k11 = {};
  v8f v00 = {}, v01 = {}, v10 = {}, v11 = {};

#pragma unroll 4
  for (int i = 0; i < KPAIR / 2; ++i) {  // uniform trip count: 128
    const v2f a0 = A0[2 * i];
    const v2f a1 = A1[2 * i];
    const int wb = (i << 10) + w0;       // i*2*DIM + half*DIM + c0
    const int wc = (i << 10) + w1;
    const v2f fq0 = Wqp[wb];
    const v2f fq1 = Wqp[wc];
    const v2f fk0 = Wkp[wb];
    const v2f fk1 = Wkp[wc];
    const v2f fv0 = Wvp[wb];
    const v2f fv1 = Wvp[wc];
    q00 = wmma_f32(a0, fq0, q00);  q01 = wmma_f32(a0, fq1, q01);
    q10 = wmma_f32(a1, fq0, q10);  q11 = wmma_f32(a1, fq1, q11);
    k00 = wmma_f32(a0, fk0, k00);  k01 = wmma_f32(a0, fk1, k01);
    k10 = wmma_f32(a1, fk0, k10);  k11 = wmma_f32(a1, fk1, k11);
    v00 = wmma_f32(a0, fv0, v00);  v01 = wmma_f32(a0, fv1, v01);
    v10 = wmma_f32(a1, fv0, v10);  v11 = wmma_f32(a1, fv1, v11);
  }

  const float bq0 = bq[c0], bq1 = bq[c1];
  const float bk0 = bk[c0], bk1 = bk[c1];
  const float bv0 = bv[c0], bv1 = bv[c1];

  // whole tile sits in one batch (32 | 4096) -> uniform padded-row shift
  const int badj = (m0 >> 12) << 2;  // +4 rows per preceding batch
  const size_t sbase = (size_t)(m0 + 8 * half + badj) * DIM + c0;
  float* Qb = Q + sbase;
  float* Kb = K + sbase;
  float* Vb = V + sbase;

#pragma unroll
  for (int r = 0; r < 8; ++r) {
    // C/D layout: VGPR r, lanes 0-15 -> M=r, lanes 16-31 -> M=8+r; N=l
    const size_t o00 = (size_t)r * DIM;          // row m0+8*half+r, col c0
    const size_t o01 = o00 + 64;                 // col c0+64 (c1 is c0+16)
    const size_t o10 = o00 + (size_t)16 * DIM;   // row +16
    const size_t o11 = o10 + 64;
    Qb[o00] = q00[r] + bq0;  Qb[o01] = q01[r] + bq1;
    Qb[o10] = q10[r] + bq0;  Qb[o11] = q11[r] + bq1;
    Kb[o00] = k00[r] + bk0;  Kb[o01] = k01[r] + bk1;
    Kb[o10] = k10[r] + bk0;  Kb[o11] = k11[r] + bk1;
    Vb[o00] = v00[r] + bv0;  Vb[o01] = v01[r] + bv1;
    Vb[o10] = v10[r] + bv0;  Vb[o11] = v11[r] + bv1;
  }
}

// ---------------------------------------------------------------------------
// 2) per-window attention. grid = (NW, NH, BATCH), block = 64.
//    Rows i=1..3 of a window are exclusive to it (plain store); rows i=0 and
//    i=4 are shared with the neighbor window (atomicAdd into zeroed accA).
// ---------------------------------------------------------------------------
__global__ __launch_bounds__(64) void attn_kernel(
    const float* __restrict__ Q, const float* __restrict__ K,
    const float* __restrict__ V, float* __restrict__ accA,
    float* __restrict__ attnW) {
  const int n = blockIdx.x, h = blockIdx.y, b = blockIdx.z;
  const int t = threadIdx.x;  // 0..63 = head dim

  __shared__ float qs[5][HD], ks[5][HD], vs[5][HD], sc[5][5];

  const size_t base = ((size_t)(b * PADP + n * 4)) * DIM + h * HD + t;
#pragma unroll
  for (int i = 0; i < 5; ++i) {
    qs[i][t] = Q[base + (size_t)i * DIM];
    ks[i][t] = K[base + (size_t)i * DIM];
    vs[i][t] = V[base + (size_t)i * DIM];
  }
  __syncthreads();

  if (t < 25) {
    const int i = t / 5, j = t % 5;
    float s = 0.0f;
#pragma unroll
    for (int d = 0; d < HD; ++d) s += qs[i][d] * ks[j][d];
    sc[i][j] = s * 0.125f;  // 1/sqrt(64)
  }
  __syncthreads();

  if (t < 5) {
    float m = sc[t][0];
#pragma unroll
    for (int j = 1; j < 5; ++j) m = fmaxf(m, sc[t][j]);
    float e[5], sum = 0.0f;
#pragma unroll
    for (int j = 0; j < 5; ++j) { e[j] = __expf(sc[t][j] - m); sum += e[j]; }
    const float inv = 1.0f / sum;
#pragma unroll
    for (int j = 0; j < 5; ++j) {
      const float a = e[j] * inv;
      sc[t][j] = a;
      atomicAdd(&attnW[((b * NH + h) * 5 + t) * 5 + j], a * (1.0f / (float)NW));
    }
  }
  __syncthreads();

#pragma unroll
  for (int i = 0; i < 5; ++i) {
    float o = 0.0f;
#pragma unroll
    for (int j = 0; j < 5; ++j) o += sc[i][j] * vs[j][t];
    float* dst = &accA[base + (size_t)i * DIM];
    if (i == 0 || i == 4) atomicAdd(dst, o);  // window-boundary overlap
    else                  *dst = o;           // exclusive row
  }
}

// ---------------------------------------------------------------------------
// 3) output projection accA @ Wo with counts/bias epilogue.
//    grid = (1024, 4), block = 128; output rows are b*S+p (contiguous).
// ---------------------------------------------------------------------------
__global__ __launch_bounds__(128) void gemm_out_kernel(
    const float* __restrict__ accA, const v2f* __restrict__ Wop,
    const float* __restrict__ bo, float* __restrict__ out) {
  const int lane = threadIdx.x & 31;
  const int wave = threadIdx.x >> 5;
  const int half = lane >> 4;
  const int l    = lane & 15;

  const int m0 = blockIdx.x * 32;
  const int n0 = blockIdx.y * 128 + wave * 32;

  const int badj = (m0 >> 12) << 2;  // uniform: tile is within one batch
  const v2f* A0 =
      (const v2f*)(accA + (size_t)(m0 + l + badj) * DIM) + half;
  const v2f* A1 =
      (const v2f*)(accA + (size_t)(m0 + 16 + l + badj) * DIM) + half;
  const int c0 = n0 + l;
  const int c1 = n0 + 16 + l;
  const int w0 = (half << 9) + c0;
  const int w1 = (half << 9) + c1;

  v8f d00 = {}, d01 = {}, d10 = {}, d11 = {};

#pragma unroll 4
  for (int i = 0; i < KPAIR / 2; ++i) {  // uniform trip count: 128
    const v2f a0 = A0[2 * i];
    const v2f a1 = A1[2 * i];
    const v2f f0 = Wop[(i << 10) + w0];
    const v2f f1 = Wop[(i << 10) + w1];
    d00 = wmma_f32(a0, f0, d00);  d01 = wmma_f32(a0, f1, d01);
    d10 = wmma_f32(a1, f0, d10);  d11 = wmma_f32(a1, f1, d11);
  }

  const float bo0 = bo[c0], bo1 = bo[c1];

  float* Ob = out + (size_t)(m0 + 8 * half) * DIM + c0;
  const int prow = (m0 + 8 * half) & (SEQ - 1);  // p of first row of subtile

#pragma unroll
  for (int r = 0; r < 8; ++r) {
#pragma unroll
    for (int s = 0; s < 2; ++s) {
      const int p = (prow + r + 16 * s) & (SEQ - 1);
      const float e0 = (s == 0) ? d00[r] : d10[r];
      const float e1 = (s == 0) ? d01[r] : d11[r];
      // interior stride boundaries (p%4==0, p>0) get 2 contributions
      const float cnt = (p != 0 && (p & 3) == 0) ? 2.0f : 1.0f;
      const float inv = 1.0f / (cnt + 1e-6f);
      const size_t o = (size_t)(r + 16 * s) * DIM;
      Ob[o]      = (e0 + cnt * bo0) * inv;
      Ob[o + 64] = (e1 + cnt * bo1) * inv;
    }
  }
}

// ---------------------------------------------------------------------------
// launch
// ---------------------------------------------------------------------------
extern "C" void kernel_launch(void* const* d_in, const int* in_sizes, int n_in,
                              void* d_out, int out_size, void* d_ws, size_t ws_size,
                              hipStream_t stream) {
  const float* x  = (const float*)d_in[0];
  const float* Wq = (const float*)d_in[1];
  const float* bq = (const float*)d_in[2];
  const float* Wk = (const float*)d_in[3];
  const float* bk = (const float*)d_in[4];
  const float* Wv = (const float*)d_in[5];
  const float* bv = (const float*)d_in[6];
  const float* Wo = (const float*)d_in[7];
  const float* bo = (const float*)d_in[8];

  float* out   = (float*)d_out;                       // (B,S,D)
  float* attnW = out + (size_t)BATCH * SEQ * DIM;     // (B,H,5,5)

  float* Q    = (float*)d_ws;
  float* K    = Q + NE;
  float* V    = K + NE;
  float* accA = V + NE;
  v2f*   Wqp  = (v2f*)(accA + NE);
  v2f*   Wkp  = Wqp + WPACK;
  v2f*   Wvp  = Wkp + WPACK;
  v2f*   Wop  = Wvp + WPACK;

  init_kernel<<<2048, 256, 0, stream>>>(accA, attnW, Q, K, V, bq, bk, bv,
                                        Wq, Wk, Wv, Wo, Wqp, Wkp, Wvp, Wop);

  gemm_qkv_kernel<<<dim3(MROWS / 32, DIM / 128), 128, 0, stream>>>(
      x, Wqp, bq, Wkp, bk, Wvp, bv, Q, K, V);

  attn_kernel<<<dim3(NW, NH, BATCH), 64, 0, stream>>>(Q, K, V, accA, attnW);

  gemm_out_kernel<<<dim3(MROWS / 32, DIM / 128), 128, 0, stream>>>(
      accA, Wop, bo, out);
}